// SortPooling_63127429317157
// MI455X (gfx1250) — compile-verified
//
#include <hip/hip_runtime.h>
#include <stdint.h>

#define K_TOP 30
#define DIM   128
#define MAXN  96
#define BLOCK 128

// Low 32 bits of a flat pointer into LDS are the LDS byte address
// (shared aperture lives in addr[63:32]).
static __device__ __forceinline__ uint32_t lds_off(const void* p) {
  return (uint32_t)(uintptr_t)p;
}

// Issue one TDM gather-mode load: 20 descriptor dwords arrive in VGPRs,
// are readfirstlane'd into the fixed SGPR window s[40:59], then
// tensor_load_to_lds consumes the four descriptor groups from there.
static __device__ __forceinline__ void tdm_gather_load(
    uint32_t a0, uint32_t a1, uint32_t a2, uint32_t a3,          // group0 (4)
    uint32_t b0, uint32_t b1, uint32_t b2, uint32_t b3,          // group1 (8)
    uint32_t b4, uint32_t b5, uint32_t b6, uint32_t b7,
    uint32_t c0, uint32_t c1, uint32_t c2, uint32_t c3,          // group2 (4)
    uint32_t d0, uint32_t d1, uint32_t d2, uint32_t d3)          // group3 (4)
{
  asm volatile(
      "v_readfirstlane_b32 s40, %0\n\t"
      "v_readfirstlane_b32 s41, %1\n\t"
      "v_readfirstlane_b32 s42, %2\n\t"
      "v_readfirstlane_b32 s43, %3\n\t"
      "v_readfirstlane_b32 s44, %4\n\t"
      "v_readfirstlane_b32 s45, %5\n\t"
      "v_readfirstlane_b32 s46, %6\n\t"
      "v_readfirstlane_b32 s47, %7\n\t"
      "v_readfirstlane_b32 s48, %8\n\t"
      "v_readfirstlane_b32 s49, %9\n\t"
      "v_readfirstlane_b32 s50, %10\n\t"
      "v_readfirstlane_b32 s51, %11\n\t"
      "v_readfirstlane_b32 s52, %12\n\t"
      "v_readfirstlane_b32 s53, %13\n\t"
      "v_readfirstlane_b32 s54, %14\n\t"
      "v_readfirstlane_b32 s55, %15\n\t"
      "v_readfirstlane_b32 s56, %16\n\t"
      "v_readfirstlane_b32 s57, %17\n\t"
      "v_readfirstlane_b32 s58, %18\n\t"
      "v_readfirstlane_b32 s59, %19\n\t"
      "s_nop 3\n\t"
      "tensor_load_to_lds s[40:43], s[44:51], s[52:55], s[56:59]"
      :
      : "v"(a0), "v"(a1), "v"(a2), "v"(a3),
        "v"(b0), "v"(b1), "v"(b2), "v"(b3),
        "v"(b4), "v"(b5), "v"(b6), "v"(b7),
        "v"(c0), "v"(c1), "v"(c2), "v"(c3),
        "v"(d0), "v"(d1), "v"(d2), "v"(d3)
      : "memory",
        "s40", "s41", "s42", "s43", "s44", "s45", "s46", "s47",
        "s48", "s49", "s50", "s51", "s52", "s53", "s54", "s55",
        "s56", "s57", "s58", "s59");
}

__global__ void __launch_bounds__(BLOCK)
sortpool_tdm_kernel(const float* __restrict__ feat,
                    const long long* __restrict__ gidx,
                    float* __restrict__ out)
{
  __shared__ float srows[K_TOP * DIM];   // gathered top-K rows (15360 B)
  __shared__ float skey[MAXN];
  __shared__ int   ssel[K_TOP];

  const int b   = blockIdx.x;
  const int tid = threadIdx.x;
  const long long start = gidx[2 * b + 0];
  const long long end   = gidx[2 * b + 1];
  int n = (int)(end - start);
  n = n < 0 ? 0 : (n > MAXN ? MAXN : n);

  // ---- phase 1: gather sort keys (last feature column) ----
  if (tid < K_TOP) ssel[tid] = -1;
  if (tid < MAXN) {
    float k = -__builtin_inff();
    if (tid < n) k = feat[(start + tid) * (long long)DIM + (DIM - 1)];
    skey[tid] = k;
  }
  __syncthreads();

  // ---- phase 2: stable descending rank == jax.lax.top_k order ----
  if (tid < MAXN) {
    const float kj = skey[tid];
    int rank = 0;
#pragma unroll 16
    for (int i = 0; i < MAXN; ++i) {
      const float ki = skey[i];
      rank += (int)((ki > kj) || ((ki == kj) && (i < tid)));
    }
    if (rank < K_TOP) ssel[rank] = (tid < n) ? tid : -1;
  }
  __syncthreads();

  // ---- phase 3: TDM gather-mode load of the 30 selected rows into LDS ----
  // One DMA per 16 indices (16-bit index mode); wave 0 issues both.
  if (tid < 32) {
    const uint64_t gbase = (uint64_t)(uintptr_t)(feat + (size_t)start * DIM);
#pragma unroll
    for (int c = 0; c < 2; ++c) {
      const int first = c * 16;
      const int nidx  = (c == 0) ? 16 : (K_TOP - 16);   // 16 then 14
      uint32_t id[16];
#pragma unroll
      for (int i = 0; i < 16; ++i) {
        const int s = (first + i < K_TOP) ? ssel[first + i] : 0;
        id[i] = (uint32_t)(s < 0 ? 0 : s);
      }
      // group0: count=1, gather_mode=1 (bit31), 16-bit indices (bit30=0)
      const uint32_t a0 = 0x80000001u;
      const uint32_t a1 = lds_off(&srows[first * DIM]);             // lds_addr
      const uint32_t a2 = (uint32_t)gbase;                          // global_addr[31:0]
      const uint32_t a3 = ((uint32_t)(gbase >> 32) & 0x01FFFFFFu) | 0x80000000u; // [56:32] | type=2
      // group1: data_size=4B (bits17:16=2); tensor_dim0=128; tensor_dim1=n;
      //         tile_dim0=128; tile_dim1=#indices; tensor_dim0_stride=128
      const uint32_t b0 = 0x00020000u;
      const uint32_t b1 = ((uint32_t)DIM & 0xFFFFu) << 16;  // tensor_dim0[15:0] @ 63:48
      const uint32_t b2 = ((uint32_t)n   & 0xFFFFu) << 16;  // tensor_dim1[15:0] @ 95:80
      const uint32_t b3 = ((uint32_t)DIM & 0xFFFFu) << 16;  // tile_dim0        @ 127:112
      const uint32_t b4 = (uint32_t)nidx;                   // tile_dim1 = #valid indices
      const uint32_t b5 = (uint32_t)DIM;                    // tensor_dim0_stride[31:0]
      const uint32_t b6 = 0u;
      const uint32_t b7 = 0u;
      // groups 2/3: packed 16-bit row indices
      tdm_gather_load(a0, a1, a2, a3,
                      b0, b1, b2, b3, b4, b5, b6, b7,
                      id[0]  | (id[1]  << 16), id[2]  | (id[3]  << 16),
                      id[4]  | (id[5]  << 16), id[6]  | (id[7]  << 16),
                      id[8]  | (id[9]  << 16), id[10] | (id[11] << 16),
                      id[12] | (id[13] << 16), id[14] | (id[15] << 16));
    }
    __builtin_amdgcn_s_wait_tensorcnt(0);
  }
  __syncthreads();

  // safety: zero rows for invalid selections (cannot occur since n >= 32 > K)
  if (tid < K_TOP && ssel[tid] < 0) {
#pragma unroll
    for (int q = 0; q < DIM; ++q) srows[tid * DIM + q] = 0.0f;
  }
  __syncthreads();

  // ---- phase 4: async store LDS tile -> out (b128, SGPR base + v offset) ----
  float* ob = out + (size_t)b * (K_TOP * DIM);
  const int NV = K_TOP * (DIM / 4);              // 960 x float4
#pragma unroll
  for (int e0 = 0; e0 < NV; e0 += BLOCK) {
    const int e = e0 + tid;
    if (e < NV) {
      const uint32_t l   = lds_off(&srows[e * 4]);
      const uint32_t off = (uint32_t)(e * 16);   // byte offset from ob
      asm volatile("global_store_async_from_lds_b128 %0, %1, %2"
                   :: "v"(off), "v"(l), "s"(ob) : "memory");
    }
  }
  asm volatile("s_wait_asynccnt 0" ::: "memory");
}

extern "C" void kernel_launch(void* const* d_in, const int* in_sizes, int n_in,
                              void* d_out, int out_size, void* d_ws, size_t ws_size,
                              hipStream_t stream) {
  (void)n_in; (void)out_size; (void)d_ws; (void)ws_size;
  const float*      feat = (const float*)d_in[0];
  const long long*  gidx = (const long long*)d_in[1];
  float*            out  = (float*)d_out;
  const int B = in_sizes[1] / 2;   // graph_indexes is (B, 2) int64
  sortpool_tdm_kernel<<<B, BLOCK, 0, stream>>>(feat, gidx, out);
}